// DPS_4277787427263
// MI455X (gfx1250) — compile-verified
//
#include <hip/hip_runtime.h>
#include <stdint.h>

// Problem constants (from reference): B=16, N=4096, K=1024, TEMP=2.0
#define B_DIM 16
#define N_DIM 4096
#define K_DIM 1024
#define INV_TEMP 0.5f

typedef float v2f __attribute__((ext_vector_type(2)));
typedef float v8f __attribute__((ext_vector_type(8)));

__device__ __forceinline__ float wave_sum(float v) {
    v += __shfl_xor(v, 16, 32);
    v += __shfl_xor(v, 8, 32);
    v += __shfl_xor(v, 4, 32);
    v += __shfl_xor(v, 2, 32);
    v += __shfl_xor(v, 1, 32);
    return v;
}

// Sum of tile totals of tiles strictly before tile (k>>8).
__device__ __forceinline__ float tile_offs(const float* tileTot, int k) {
    int q = k >> 8;
    float o = 0.0f;
    if (q > 0) o += tileTot[0];
    if (q > 1) o += tileTot[1];
    if (q > 2) o += tileTot[2];
    return o;
}

// Inclusive prefix scan of in[0..1023] -> out[0..1023] (per-tile; caller adds
// tile_offs). Four waves each scan one 256-element tile with WMMA:
//   W = L x V        (per-16-segment inclusive scans; L lower-triangular ones)
//   S = W + T x U'   (T = broadcast segment totals, U' strictly-upper ones)
// Each 16x16x16 f32 product = 4 chained V_WMMA_F32_16X16X4_F32.
__device__ __forceinline__ void scan1024(const float* in, float* out,
                                         float* segTot, float* tileTot, int t) {
    __syncthreads();   // publish `in` (and anything written just before)
    const int wav = t >> 5;
    if (wav < 4) {
        const int l    = t & 31;
        const int half = l >> 4;     // lane group: 0 -> lanes 0-15, 1 -> 16-31
        const int m    = l & 15;     // row (A) and column (B/D) index
        const int base = wav * 256;
        v8f c = {};
        // ---- W = L x V ----
#pragma unroll
        for (int tk = 0; tk < 4; ++tk) {
            const int k0 = 4 * tk + 2 * half;   // local K of vgpr0 slot
            v2f a, bb;
            a.x  = (k0     <= m) ? 1.0f : 0.0f;  // L[m][k0]
            a.y  = (k0 + 1 <= m) ? 1.0f : 0.0f;  // L[m][k0+1]
            bb.x = in[base + m * 16 + k0];       // V[k0][m]
            bb.y = in[base + m * 16 + k0 + 1];   // V[k0+1][m]
            c = __builtin_amdgcn_wmma_f32_16x16x4_f32(
                    false, a, false, bb, (short)0, c, false, false);
        }
        // segment totals: W[15][n] lives in c[7] of lanes 16-31 (M=15)
        if (half) segTot[wav * 16 + m] = c[7];
        asm volatile("s_wait_dscnt 0" ::: "memory");
        // ---- S = W + T x Ustrict ----
#pragma unroll
        for (int tk = 0; tk < 4; ++tk) {
            const int k0 = 4 * tk + 2 * half;
            v2f a, bb;
            a.x  = segTot[wav * 16 + k0];        // T[m][k0]   = tot[k0]
            a.y  = segTot[wav * 16 + k0 + 1];    // T[m][k0+1] = tot[k0+1]
            bb.x = (k0     < m) ? 1.0f : 0.0f;   // Ustrict[k0][m]
            bb.y = (k0 + 1 < m) ? 1.0f : 0.0f;   // Ustrict[k0+1][m]
            c = __builtin_amdgcn_wmma_f32_16x16x4_f32(
                    false, a, false, bb, (short)0, c, false, false);
        }
        // D layout: c[p] holds element (M = p + 8*half, N = m); tile index = N*16 + M
#pragma unroll
        for (int p = 0; p < 8; ++p)
            out[base + m * 16 + p + 8 * half] = c[p];
        if (l == 31) tileTot[wav] = c[7];        // element 255 of tile
    }
    __syncthreads();
}

__global__ __launch_bounds__(1024)
void gumbel_topk_st_kernel(const float* __restrict__ x,
                           const float* __restrict__ logits,
                           const float* __restrict__ noise,
                           float* __restrict__ out) {
    __shared__ float    sv[N_DIM];       // keys (base values), later sorted desc
    __shared__ unsigned si[N_DIM];       // payload (orig index); logits staging first
    __shared__ float    scanA[K_DIM];    // scan input
    __shared__ float    scanB[K_DIM];    // scan output
    __shared__ float    segTot[64];      // 16 segment totals x 4 tiles
    __shared__ float    tileTot[4];
    __shared__ float    wavePart[32];
    __shared__ float    redS[1];

    const int t    = threadIdx.x;
    const int b    = blockIdx.x;
    const int lane = t & 31;
    const int wav  = t >> 5;
    const size_t rowOff = (size_t)b * N_DIM;

    // warm L2 for the x row we gather at the end
    __builtin_prefetch(x + rowOff + (size_t)t * 4, 0, 1);

    // ---- Phase 0: async-copy noise row -> sv, logits -> si (raw f32 bytes) ----
    {
        unsigned ldsN = (unsigned)(uintptr_t)(&sv[0]) + (unsigned)t * 16u;
        unsigned ldsL = (unsigned)(uintptr_t)(&si[0]) + (unsigned)t * 16u;
        unsigned long long gN = (unsigned long long)(uintptr_t)(noise + rowOff)
                                + (unsigned long long)t * 16ull;
        unsigned long long gL = (unsigned long long)(uintptr_t)(logits)
                                + (unsigned long long)t * 16ull;
        asm volatile("global_load_async_to_lds_b128 %0, %1, off"
                     :: "v"(ldsN), "v"(gN) : "memory");
        asm volatile("global_load_async_to_lds_b128 %0, %1, off"
                     :: "v"(ldsL), "v"(gL) : "memory");
        asm volatile("s_wait_asynccnt 0" ::: "memory");
    }
    __syncthreads();

    // ---- Phase 1: base = logits + noise; attach original index ----
#pragma unroll
    for (int i = 0; i < 4; ++i) {
        int n = t + i * 1024;
        float lg = __uint_as_float(si[n]);   // logits staged as raw bits
        float nz = sv[n];
        sv[n] = lg + nz;
        si[n] = (unsigned)n;
    }

    // ---- Phase 2: bitonic sort, descending by value (ties: smaller index) ----
    for (unsigned size = 2; size <= (unsigned)N_DIM; size <<= 1) {
        for (unsigned stride = size >> 1; stride >= 1; stride >>= 1) {
            __syncthreads();
#pragma unroll
            for (int vv = 0; vv < 2; ++vv) {
                unsigned p = (unsigned)t + (unsigned)vv * 1024u;
                unsigned i = 2u * p - (p & (stride - 1u));
                unsigned j = i + stride;
                float    fi = sv[i], fj = sv[j];
                unsigned ii = si[i], ij = si[j];
                bool jFirst    = (fj > fi) || (fj == fi && ij < ii);
                bool descBlock = ((i & size) == 0u);
                if (jFirst == descBlock) {
                    sv[i] = fj; sv[j] = fi;
                    si[i] = ij; si[j] = ii;
                }
            }
        }
    }
    __syncthreads();

    // ---- Phase 3: exponentials (max-shifted) + total sum S over all 4096 ----
    const float M = sv[0];                 // sorted descending -> row max
    float local = 0.0f, e0 = 0.0f;
#pragma unroll
    for (int i = 0; i < 4; ++i) {
        int j = t + i * 1024;
        float e = __expf((sv[j] - M) * INV_TEMP);
        if (i == 0) e0 = e;
        local += e;
    }
    scanA[t] = e0;                         // e of sorted rank t (t < 1024)
    float ws = wave_sum(local);
    if (lane == 0) wavePart[wav] = ws;
    __syncthreads();
    if (t < 32) {
        float v = wave_sum(wavePart[t]);
        if (t == 0) redS[0] = v;
    }

    // ---- Phase 4: inclusive scan of e_sorted (WMMA) -> P_k = incl_k - e_k ----
    scan1024(scanA, scanB, segTot, tileTot, t);
    const float S = redS[0];
    {
        float incl = scanB[t] + tile_offs(tileTot, t);
        float P    = incl - scanA[t];      // exclusive prefix of e_sorted
        float D    = S - P;                // column-k softmax denominator
        scanA[t]   = 1.0f / D;             // reciprocal denominators r_k
    }

    // ---- Phase 5: inclusive scan of r_k (WMMA) -> Cinc, Ctotal ----
    scan1024(scanA, scanB, segTot, tileTot, t);
    const float Ctot = scanB[K_DIM - 1] + tileTot[0] + tileTot[1] + tileTot[2];

    // ---- Phase 6: scatter straight-through output by original index ----
#pragma unroll
    for (int i = 0; i < 4; ++i) {
        int j = t + i * 1024;
        unsigned n = si[j];
        float e = __expf((sv[j] - M) * INV_TEMP);
        float cfac, hard;
        if (i == 0) {                      // j < 1024: selected at rank j
            cfac = scanB[j] + tile_offs(tileTot, j);
            hard = 1.0f;
        } else {                           // not selected
            cfac = Ctot;
            hard = 0.0f;
        }
        float coll = e * cfac;
        float samp = (hard - coll) + coll; // straight-through forward
        out[rowOff + n] = samp * x[rowOff + n];
    }
}

extern "C" void kernel_launch(void* const* d_in, const int* in_sizes, int n_in,
                              void* d_out, int out_size, void* d_ws, size_t ws_size,
                              hipStream_t stream) {
    const float* x      = (const float*)d_in[0];
    const float* logits = (const float*)d_in[1];
    const float* noise  = (const float*)d_in[2];
    float* out = (float*)d_out;
    (void)in_sizes; (void)n_in; (void)out_size; (void)d_ws; (void)ws_size;
    gumbel_topk_st_kernel<<<dim3(B_DIM), dim3(1024), 0, stream>>>(x, logits, noise, out);
}